// RotateNMS_54571854463598
// MI455X (gfx1250) — compile-verified
//
#include <hip/hip_runtime.h>
#include <hip/hip_bf16.h>

typedef __attribute__((ext_vector_type(2))) float v2f;
typedef __attribute__((ext_vector_type(8))) float v8f;

#define NMS_THR 0.7f

// ---------------------------------------------------------------------------
// Kernel 1: per-box prep: CCW corners, center, |c|^2, circumradius, area
// ---------------------------------------------------------------------------
__global__ __launch_bounds__(256) void rnms_prep(
    const float* __restrict__ boxes,  // (N,5) xc,yc,w,h,theta_deg
    float* __restrict__ cor,          // (Npad,8) x0,y0,...,x3,y3
    float* __restrict__ cx, float* __restrict__ cy, float* __restrict__ cs,
    float* __restrict__ rad, float* __restrict__ area, int N) {
  int i = blockIdx.x * blockDim.x + threadIdx.x;
  if (i >= N) return;
  float xc = boxes[i * 5 + 0];
  float yc = boxes[i * 5 + 1];
  float w  = boxes[i * 5 + 2];
  float h  = boxes[i * 5 + 3];
  float t  = boxes[i * 5 + 4];
  float th = t * 0.017453292519943295f;
  float c = __cosf(th), s = __sinf(th);
  // use precise sin/cos to match reference closely
  c = cosf(th); s = sinf(th);
  const float lx[4] = {0.5f, -0.5f, -0.5f, 0.5f};
  const float ly[4] = {0.5f, 0.5f, -0.5f, -0.5f};
#pragma unroll
  for (int k = 0; k < 4; ++k) {
    float x = xc + lx[k] * w * c - ly[k] * h * s;
    float y = yc + lx[k] * w * s + ly[k] * h * c;
    cor[i * 8 + 2 * k]     = x;
    cor[i * 8 + 2 * k + 1] = y;
  }
  cx[i] = xc;
  cy[i] = yc;
  cs[i] = xc * xc + yc * yc;
  rad[i] = 0.5f * sqrtf(w * w + h * h);
  area[i] = w * h;
}

// ---------------------------------------------------------------------------
// Exact convex-quad intersection area (Sutherland–Hodgman, CCW quads)
// ---------------------------------------------------------------------------
__device__ inline float quadInterArea(const float* __restrict__ A,
                                      const float* __restrict__ B) {
  float px[8], py[8], qx[8], qy[8];
  int n = 4;
#pragma unroll
  for (int k = 0; k < 4; ++k) { px[k] = A[2 * k]; py[k] = A[2 * k + 1]; }
#pragma unroll
  for (int e = 0; e < 4; ++e) {
    float cxe = B[2 * e], cye = B[2 * e + 1];
    int e1 = (e + 1) & 3;
    float ex = B[2 * e1] - cxe, ey = B[2 * e1 + 1] - cye;
    int m = 0;
    for (int k = 0; k < n; ++k) {
      int kk = (k + 1 == n) ? 0 : k + 1;
      float x0 = px[k], y0 = py[k];
      float x1 = px[kk], y1 = py[kk];
      float s0 = ex * (y0 - cye) - ey * (x0 - cxe);
      float s1 = ex * (y1 - cye) - ey * (x1 - cxe);
      bool in0 = s0 >= 0.0f, in1 = s1 >= 0.0f;
      if (in0) { qx[m] = x0; qy[m] = y0; ++m; }
      if (in0 != in1) {
        float t = s0 / (s0 - s1);
        qx[m] = x0 + t * (x1 - x0);
        qy[m] = y0 + t * (y1 - y0);
        ++m;
      }
    }
    n = m;
    if (n < 3) return 0.0f;
    for (int k = 0; k < n; ++k) { px[k] = qx[k]; py[k] = qy[k]; }
  }
  float area2 = 0.0f;
  for (int k = 0; k < n; ++k) {
    int kk = (k + 1 == n) ? 0 : k + 1;
    area2 += px[k] * py[kk] - px[kk] * py[k];
  }
  return 0.5f * fabsf(area2);
}

// ---------------------------------------------------------------------------
// Kernel 2: one wave32 per 16x16 pair tile.
//   WMMA_F32_16X16X4_F32 computes dist^2(i,j) for the whole tile:
//     A row m = [x_i, y_i, |c_i|^2, 1], B col n = [-2x_j, -2y_j, 1, |c_j|^2]
//   Survivors of the circumradius prefilter get exact polygon-clip IoU.
//   Suppression bits ballot-packed into exclusive 16-bit mask halfwords.
// ---------------------------------------------------------------------------
__global__ __launch_bounds__(32) void rnms_pair_tiles(
    const float* __restrict__ cor, const float* __restrict__ cx,
    const float* __restrict__ cy, const float* __restrict__ cs,
    const float* __restrict__ rad, const float* __restrict__ area,
    unsigned short* __restrict__ maskHalf, int N, int T) {
  int tj = blockIdx.x;  // column tile
  int ti = blockIdx.y;  // row tile
  if (tj < ti) return;  // lower triangle left unwritten; greedy pass still reads it
  int lane = threadIdx.x;
  int t = lane & 15;

  __shared__ float rCor[16 * 8], cCor[16 * 8];
  __shared__ float rRad[16], cRad[16], rArea[16], cArea[16];

  // Cooperative LDS fill: lanes 0-15 load row boxes, lanes 16-31 col boxes.
  {
    int half = lane >> 4;
    int box = (half ? tj : ti) * 16 + t;
    int bc = box < N ? box : N - 1;
    float* dstC = half ? cCor : rCor;
    float* dstR = half ? cRad : rRad;
    float* dstA = half ? cArea : rArea;
#pragma unroll
    for (int k = 0; k < 8; ++k) dstC[t * 8 + k] = cor[bc * 8 + k];
    dstR[t] = rad[bc];
    dstA[t] = area[bc];
  }

  // Branchless WMMA fragment build (EXEC all-ones at the WMMA).
  int iRow = ti * 16 + t; int iR = iRow < N ? iRow : N - 1;
  int jCol = tj * 16 + t; int jC = jCol < N ? jCol : N - 1;
  bool hi = lane >= 16;
  float axv = cx[iR], ayv = cy[iR], asv = cs[iR];
  float bxv = cx[jC], byv = cy[jC], bsv = cs[jC];
  v2f a, b;
  a.x = hi ? asv : axv;            // K=0 -> x_i        | K=2 -> |c_i|^2
  a.y = hi ? 1.0f : ayv;           // K=1 -> y_i        | K=3 -> 1
  b.x = hi ? 1.0f : -2.0f * bxv;   // K=0 -> -2 x_j     | K=2 -> 1
  b.y = hi ? bsv : -2.0f * byv;    // K=1 -> -2 y_j     | K=3 -> |c_j|^2
  v8f acc = {};
  v8f d = __builtin_amdgcn_wmma_f32_16x16x4_f32(
      /*neg_a=*/false, a, /*neg_b=*/false, b,
      /*c_mod=*/(short)0, acc, /*reuse_a=*/false, /*reuse_b=*/false);

  __syncthreads();

#pragma unroll
  for (int r = 0; r < 8; ++r) {
    int m = hi ? r + 8 : r;   // D VGPR r holds rows r (lanes 0-15) and r+8
    int n = t;                // column within tile = lane & 15
    int gi = ti * 16 + m;
    int gj = tj * 16 + n;
    int pred = 0;
    if (gi < N && gj < N && gj > gi) {
      float d2 = d[r];
      float rr = rRad[m] + cRad[n];
      if (d2 <= rr * rr + 4.0f) {  // conservative circumradius prefilter
        float inter = quadInterArea(&rCor[m * 8], &cCor[n * 8]);
        float uni = rArea[m] + cArea[n] - inter;
        float iou = inter / fmaxf(uni, 1e-8f);
        pred = (iou > NMS_THR) ? 1 : 0;
      }
    }
    unsigned bal = (unsigned)__ballot(pred);  // wave32: bit = lane
    if (lane == 0)
      maskHalf[(size_t)(ti * 16 + r) * T + tj] = (unsigned short)(bal & 0xFFFFu);
    if (lane == 16)
      maskHalf[(size_t)(ti * 16 + r + 8) * T + tj] = (unsigned short)(bal >> 16);
  }
}

// ---------------------------------------------------------------------------
// Kernel 3: single-block greedy suppression scan + compaction.
// ---------------------------------------------------------------------------
__global__ __launch_bounds__(256) void rnms_greedy(
    const unsigned short* __restrict__ maskHalf, int N, int T,
    int* __restrict__ out, int out_size) {
  __shared__ unsigned char removed[4096];
  int tid = threadIdx.x;
  for (int k = tid; k < N; k += 256) removed[k] = 0;
  __syncthreads();
  for (int i = 0; i < N; ++i) {
    __syncthreads();
    if (removed[i]) continue;  // uniform LDS broadcast
    const unsigned short* row = maskHalf + (size_t)i * T;
    for (int hw = tid; hw < T; hw += 256) {
      unsigned m = row[hw];
      while (m) {
        int bpos = __ffs(m) - 1;
        m &= m - 1;
        int j = hw * 16 + bpos;  // construction guarantees i < j < N
        removed[j] = 1;
      }
    }
  }
  __syncthreads();
  if (tid == 0) {
    int cnt = 0;
    for (int i = 0; i < N && cnt < out_size; ++i)
      if (!removed[i]) out[cnt++] = i;
    for (; cnt < out_size; ++cnt) out[cnt] = -1;
  }
}

// ---------------------------------------------------------------------------
extern "C" void kernel_launch(void* const* d_in, const int* in_sizes, int n_in,
                              void* d_out, int out_size, void* d_ws, size_t ws_size,
                              hipStream_t stream) {
  (void)n_in; (void)ws_size;
  const float* boxes = (const float*)d_in[0];
  int N = in_sizes[0] / 5;
  int T = (N + 15) / 16;
  int Npad = T * 16;

  char* ws = (char*)d_ws;
  float* cor  = (float*)ws; ws += (size_t)Npad * 8 * sizeof(float);
  float* cx   = (float*)ws; ws += (size_t)Npad * sizeof(float);
  float* cy   = (float*)ws; ws += (size_t)Npad * sizeof(float);
  float* cs   = (float*)ws; ws += (size_t)Npad * sizeof(float);
  float* rad  = (float*)ws; ws += (size_t)Npad * sizeof(float);
  float* area = (float*)ws; ws += (size_t)Npad * sizeof(float);
  unsigned short* maskHalf = (unsigned short*)ws;  // Npad * T halfwords

  int* out = (int*)d_out;

  rnms_prep<<<(N + 255) / 256, 256, 0, stream>>>(boxes, cor, cx, cy, cs, rad,
                                                 area, N);
  dim3 grid(T, T);
  rnms_pair_tiles<<<grid, 32, 0, stream>>>(cor, cx, cy, cs, rad, area,
                                           maskHalf, N, T);
  rnms_greedy<<<1, 256, 0, stream>>>(maskHalf, N, T, out, out_size);
}